// AttentionPooling_47794396070379
// MI455X (gfx1250) — compile-verified
//
#include <hip/hip_runtime.h>
#include <hip/hip_bf16.h>

typedef __attribute__((ext_vector_type(2))) float v2f;
typedef __attribute__((ext_vector_type(4))) float v4f;
typedef __attribute__((ext_vector_type(8))) float v8f;

#define LANES 32
#define D 128

// ---------------------------------------------------------------------------
// Pass 1: per-segment sums of x and row counts.
// One wave per contiguous chunk of rows; lane l owns columns [4l, 4l+3].
// Register accumulation within a segment run; atomic commit at boundaries.
// x is streamed with non-temporal loads (read-once, 1 GB) so csum/cnt/batch
// stay resident in the 192 MB L2.
// ---------------------------------------------------------------------------
__global__ __launch_bounds__(256) void seg_sum_kernel(
    const float* __restrict__ x, const int* __restrict__ batch,
    float* __restrict__ csum, float* __restrict__ cnt,
    int N, int rows_per_wave) {
  const int wave = (blockIdx.x * blockDim.x + threadIdx.x) >> 5;
  const int lane = threadIdx.x & (LANES - 1);
  long start = (long)wave * rows_per_wave;
  if (start >= N) return;
  long end = start + rows_per_wave;
  if (end > N) end = N;
  const int col = lane * 4;

  v4f acc = {0.f, 0.f, 0.f, 0.f};
  float cc = 0.f;
  int cur = -1;

  for (long r = start; r < end; ++r) {
    // issue the streaming load before the (rare) boundary branch
    const v4f xv = __builtin_nontemporal_load((const v4f*)(x + (size_t)r * D + col));
    const int seg = batch[r];
    if (seg != cur) {
      if (cur >= 0) {
        float* p = csum + (size_t)cur * D + col;
        atomicAdd(p + 0, acc[0]);
        atomicAdd(p + 1, acc[1]);
        atomicAdd(p + 2, acc[2]);
        atomicAdd(p + 3, acc[3]);
        if (lane == 0) atomicAdd(cnt + cur, cc);
      }
      acc = (v4f){0.f, 0.f, 0.f, 0.f};
      cc = 0.f;
      cur = seg;
    }
    acc += xv;
    cc += 1.f;
  }
  if (cur >= 0) {
    float* p = csum + (size_t)cur * D + col;
    atomicAdd(p + 0, acc[0]);
    atomicAdd(p + 1, acc[1]);
    atomicAdd(p + 2, acc[2]);
    atomicAdd(p + 3, acc[3]);
    if (lane == 0) atomicAdd(cnt + cur, cc);
  }
}

// ---------------------------------------------------------------------------
// Pass 2: att[i] = <x_i, coarse[batch_i]>, accumulate att*x per segment.
// coarse row (csum/count) is loaded once per segment run and held in regs.
// Per-row dot: 4 FMAs/lane + 5-step wave32 xor reduction.
// ---------------------------------------------------------------------------
__global__ __launch_bounds__(256) void att_pool_kernel(
    const float* __restrict__ x, const int* __restrict__ batch,
    const float* __restrict__ csum, const float* __restrict__ cnt,
    float* __restrict__ psum, int N, int rows_per_wave) {
  const int wave = (blockIdx.x * blockDim.x + threadIdx.x) >> 5;
  const int lane = threadIdx.x & (LANES - 1);
  long start = (long)wave * rows_per_wave;
  if (start >= N) return;
  long end = start + rows_per_wave;
  if (end > N) end = N;
  const int col = lane * 4;

  v4f acc = {0.f, 0.f, 0.f, 0.f};
  v4f cv  = {0.f, 0.f, 0.f, 0.f};
  int cur = -1;

  for (long r = start; r < end; ++r) {
    const v4f xv = __builtin_nontemporal_load((const v4f*)(x + (size_t)r * D + col));
    const int seg = batch[r];
    if (seg != cur) {
      if (cur >= 0) {
        float* p = psum + (size_t)cur * D + col;
        atomicAdd(p + 0, acc[0]);
        atomicAdd(p + 1, acc[1]);
        atomicAdd(p + 2, acc[2]);
        atomicAdd(p + 3, acc[3]);
      }
      acc = (v4f){0.f, 0.f, 0.f, 0.f};
      const float inv = 1.f / fmaxf(cnt[seg], 1.f);
      const v4f cs = *(const v4f*)(csum + (size_t)seg * D + col);
      cv = cs * inv;
      cur = seg;
    }
    float p = xv[0] * cv[0] + xv[1] * cv[1] + xv[2] * cv[2] + xv[3] * cv[3];
#pragma unroll
    for (int off = 16; off > 0; off >>= 1) p += __shfl_xor(p, off, LANES);
    acc += p * xv;
  }
  if (cur >= 0) {
    float* p = psum + (size_t)cur * D + col;
    atomicAdd(p + 0, acc[0]);
    atomicAdd(p + 1, acc[1]);
    atomicAdd(p + 2, acc[2]);
    atomicAdd(p + 3, acc[3]);
  }
}

// ---------------------------------------------------------------------------
// Final GEMM: out[m][n] = sum_k (psum[m][k]/cnt[m]) * W[n][k] + b[n]
// One wave per 16x16 output tile; K=128 in steps of 4 via v_wmma_f32_16x16x4_f32
// (full fp32 precision). Count division fused into A loads; bias pre-loaded
// into the C accumulator. Output stores are non-temporal (write-once).
// A (16x4 f32): lanes 0-15 row M=lane {K=k0,k0+1}; lanes 16-31 {K=k0+2,k0+3}.
// B (4x16 f32): lanes 0-15 col N=lane {K=k0,k0+1}; lanes 16-31 {K=k0+2,k0+3}.
// C/D: VGPR j -> M=j (lanes 0-15) / M=j+8 (lanes 16-31), N=lane&15.
// ---------------------------------------------------------------------------
__global__ __launch_bounds__(256) void gemm_wmma_kernel(
    const float* __restrict__ psum, const float* __restrict__ cnt,
    const float* __restrict__ W, const float* __restrict__ bias,
    float* __restrict__ out, int B) {
  const int wave = (blockIdx.x * blockDim.x + threadIdx.x) >> 5;
  const int lane = threadIdx.x & (LANES - 1);
  const int nTilesN = D / 16;               // 8
  const int tileN = wave & (nTilesN - 1);
  const int tileM = wave >> 3;
  if (tileM * 16 >= B) return;

  const int hi = lane >> 4;                 // 0 or 1
  const int lo = lane & 15;
  const int rowA = tileM * 16 + lo;         // A-matrix row for this lane
  const int ncol = tileN * 16 + lo;         // B-matrix column for this lane
  const float invc = 1.f / fmaxf(cnt[rowA], 1.f);

  v8f c;
  const float bv = bias[ncol];
#pragma unroll
  for (int j = 0; j < 8; ++j) c[j] = bv;

  const float* arow = psum + (size_t)rowA * D;
  const float* brow = W + (size_t)ncol * D;

#pragma unroll 4
  for (int k0 = 0; k0 < D; k0 += 4) {
    const int kk = k0 + 2 * hi;
    v2f a, bb;
    a[0] = arow[kk] * invc;
    a[1] = arow[kk + 1] * invc;
    bb[0] = brow[kk];
    bb[1] = brow[kk + 1];
    c = __builtin_amdgcn_wmma_f32_16x16x4_f32(
        /*neg_a=*/false, a, /*neg_b=*/false, bb,
        /*c_mod=*/(short)0, c, /*reuse_a=*/false, /*reuse_b=*/false);
  }

#pragma unroll
  for (int j = 0; j < 8; ++j) {
    __builtin_nontemporal_store(
        c[j], out + (size_t)(tileM * 16 + j + 8 * hi) * D + ncol);
  }
}

// ---------------------------------------------------------------------------
extern "C" void kernel_launch(void* const* d_in, const int* in_sizes, int n_in,
                              void* d_out, int out_size, void* d_ws, size_t ws_size,
                              hipStream_t stream) {
  const float* x     = (const float*)d_in[0];
  const int*   batch = (const int*)d_in[1];
  const float* W     = (const float*)d_in[2];
  const float* bias  = (const float*)d_in[3];

  const int N = in_sizes[1];          // 2,000,000 rows
  const int B = out_size / D;         // 8192 segments

  float* csum = (float*)d_ws;                      // [B,128]
  float* psum = csum + (size_t)B * D;              // [B,128]
  float* cnt  = psum + (size_t)B * D;              // [B]
  const size_t zero_bytes = ((size_t)B * D * 2 + (size_t)B) * sizeof(float);
  hipMemsetAsync(d_ws, 0, zero_bytes, stream);

  // Pass 1 + Pass 2: 8192 waves, contiguous row chunks (~245 rows each).
  const int waves = 8192;
  const int rows_per_wave = (N + waves - 1) / waves;
  const int blocks = waves / 8;       // 256 threads = 8 waves per block
  seg_sum_kernel<<<blocks, 256, 0, stream>>>(x, batch, csum, cnt, N, rows_per_wave);
  att_pool_kernel<<<blocks, 256, 0, stream>>>(x, batch, csum, cnt, psum, N, rows_per_wave);

  // GEMM: (B/16)*(128/16) tiles, one wave each.
  const int tiles = (B / 16) * (D / 16);
  gemm_wmma_kernel<<<tiles / 8, 256, 0, stream>>>(psum, cnt, W, bias, (float*)d_out, B);
}